// GATModel_3779571221055
// MI455X (gfx1250) — compile-verified
//
#include <hip/hip_runtime.h>
#include <hip/hip_bf16.h>
#include <math.h>

typedef __attribute__((ext_vector_type(2))) float v2f;
typedef __attribute__((ext_vector_type(8))) float v8f;

// ---------------------------------------------------------------------------
// fp32 WMMA GEMM: C[M,N] = A[M,K] @ B[K,N] (+ optional bias)
// One wave computes one 16x16 tile via V_WMMA_F32_16X16X4_F32, K stepped by 4.
// A frag (16x4): lane L -> m=L&15, holds K = 2*(L>>4) .. +1 in 2 VGPRs.
// B frag (4x16): lane L -> n=L&15, holds K = 2*(L>>4) .. +1 in 2 VGPRs.
// C/D (16x16):   lane L -> n=L&15, VGPR i -> m = i + 8*(L>>4).
// ---------------------------------------------------------------------------
template <bool ADD_BIAS>
__global__ void gemm_f32_wmma(const float* __restrict__ A,
                              const float* __restrict__ B,
                              const float* __restrict__ bias,
                              float* __restrict__ C,
                              int M, int N, int K) {
    const int wave = threadIdx.x >> 5;
    const int lane = threadIdx.x & 31;
    const int ntn  = N >> 4;                 // number of 16-wide col tiles
    const int ct   = blockIdx.y * 4 + wave;  // col tile for this wave
    if (ct >= ntn) return;                   // whole wave exits -> EXEC stays full
    const int rt = blockIdx.x;
    const int m0 = rt << 4;
    const int n0 = ct << 4;

    const int mrow  = lane & 15;
    const int khalf = (lane >> 4) << 1;      // 0 or 2

    // clamp row for safety (M divides 16 in this problem, but be robust)
    const int arow = (m0 + mrow < M) ? (m0 + mrow) : (M - 1);

    v8f acc = {};
    for (int k0 = 0; k0 < K; k0 += 4) {
        v2f a, b;
        const float* ap = A + (size_t)arow * K + (k0 + khalf);
        a[0] = ap[0];
        a[1] = ap[1];
        const float* bp = B + (size_t)(k0 + khalf) * N + (n0 + mrow);
        b[0] = bp[0];
        b[1] = bp[N];
        acc = __builtin_amdgcn_wmma_f32_16x16x4_f32(
            /*neg_a=*/false, a, /*neg_b=*/false, b,
            /*c_mod=*/(short)0, acc, /*reuse_a=*/false, /*reuse_b=*/false);
    }

    const int n     = n0 + (lane & 15);
    const int mbase = m0 + ((lane >> 4) << 3);
    const float bv  = ADD_BIAS ? bias[n] : 0.0f;
#pragma unroll
    for (int i = 0; i < 8; ++i) {
        const int m = mbase + i;
        if (m < M) C[(size_t)m * N + n] = acc[i] + bv;
    }
}

// ---------------------------------------------------------------------------
// Per-(node,head) attention scores: asrc[n,h]=<h[n,h,:],att_s[h,:]>, same for dst
// ---------------------------------------------------------------------------
__global__ void att_scores(const float* __restrict__ h,
                           const float* __restrict__ att_s,
                           const float* __restrict__ att_d,
                           float* __restrict__ asrc, float* __restrict__ adst,
                           int N, int H, int C) {
    int t = blockIdx.x * blockDim.x + threadIdx.x;
    if (t >= N * H) return;
    int n = t / H, hh = t - n * H;
    const float* hp = h + (size_t)n * H * C + (size_t)hh * C;
    const float* as = att_s + (size_t)hh * C;
    const float* ad = att_d + (size_t)hh * C;
    float ss = 0.f, sd = 0.f;
    for (int c = 0; c < C; ++c) {
        float v = hp[c];
        ss += v * as[c];
        sd += v * ad[c];
    }
    asrc[t] = ss;
    adst[t] = sd;
}

__global__ void fill_f32(float* __restrict__ p, float v, int n) {
    int t = blockIdx.x * blockDim.x + threadIdx.x;
    if (t < n) p[t] = v;
}

__device__ __forceinline__ float leaky(float v) {
    return v > 0.f ? v : 0.2f * v;
}

// float atomic max via int/uint trick; dst initialized to -inf (0xFF800000)
__device__ __forceinline__ void atomicMaxF(float* addr, float val) {
    if (val >= 0.f)
        atomicMax((int*)addr, __float_as_int(val));
    else
        atomicMin((unsigned int*)addr, __float_as_uint(val));
}

// ---------------------------------------------------------------------------
// Edge pass 1: segment max of leaky_relu(asrc[src]+adst[dst]) over dst
// Edges 0..E-1 come from edge_index; E..E+N-1 are implicit self loops.
// ---------------------------------------------------------------------------
__global__ void edge_max(const int* __restrict__ ei, int E, int N, int H,
                         const float* __restrict__ asrc,
                         const float* __restrict__ adst,
                         float* __restrict__ nmax) {
    int e = blockIdx.x * blockDim.x + threadIdx.x;
    if (e >= E + N) return;
    int s, d;
    if (e < E) { s = ei[e]; d = ei[E + e]; } else { s = d = e - E; }
    for (int h = 0; h < H; ++h) {
        float v = leaky(asrc[(size_t)s * H + h] + adst[(size_t)d * H + h]);
        atomicMaxF(&nmax[(size_t)d * H + h], v);
    }
}

// Edge pass 2: denom[dst,h] += exp(e - max[dst,h])
__global__ void edge_expsum(const int* __restrict__ ei, int E, int N, int H,
                            const float* __restrict__ asrc,
                            const float* __restrict__ adst,
                            const float* __restrict__ nmax,
                            float* __restrict__ den) {
    int e = blockIdx.x * blockDim.x + threadIdx.x;
    if (e >= E + N) return;
    int s, d;
    if (e < E) { s = ei[e]; d = ei[E + e]; } else { s = d = e - E; }
    for (int h = 0; h < H; ++h) {
        float v  = leaky(asrc[(size_t)s * H + h] + adst[(size_t)d * H + h]);
        float ex = __expf(v - nmax[(size_t)d * H + h]);
        atomicAdd(&den[(size_t)d * H + h], ex);
    }
}

// Edge pass 3: agg[dst, h, c] += h[src, h, c] * alpha[e, h]
// blockDim.x = 256; each block handles 256/(H*C) edges; alpha recomputed.
__global__ void edge_aggregate(const int* __restrict__ ei, int E, int N,
                               int H, int C,
                               const float* __restrict__ hmat,
                               const float* __restrict__ asrc,
                               const float* __restrict__ adst,
                               const float* __restrict__ nmax,
                               const float* __restrict__ den,
                               float* __restrict__ agg) {
    const int HC  = H * C;
    const int epb = blockDim.x / HC;
    const int li  = threadIdx.x;
    const int el  = li / HC;
    const int j   = li - el * HC;
    long long e   = (long long)blockIdx.x * epb + el;
    if (e >= (long long)E + N) return;
    int s, d;
    if (e < E) { s = ei[e]; d = ei[E + e]; } else { s = d = (int)(e - E); }
    const int hh = j / C;
    float v     = leaky(asrc[(size_t)s * H + hh] + adst[(size_t)d * H + hh]);
    float alpha = __expf(v - nmax[(size_t)d * H + hh]) /
                  (den[(size_t)d * H + hh] + 1e-16f);
    atomicAdd(&agg[(size_t)d * HC + j], hmat[(size_t)s * HC + j] * alpha);
}

// out = elu(agg + bias), in place
__global__ void bias_elu(float* __restrict__ a, const float* __restrict__ bias,
                         int total, int HC) {
    int t = blockIdx.x * blockDim.x + threadIdx.x;
    if (t >= total) return;
    float v = a[t] + bias[t % HC];
    a[t] = v > 0.f ? v : (__expf(v) - 1.0f);
}

// ---------------------------------------------------------------------------
// Launch: full 2-layer GAT + output projection
// ---------------------------------------------------------------------------
extern "C" void kernel_launch(void* const* d_in, const int* in_sizes, int n_in,
                              void* d_out, int out_size, void* d_ws, size_t ws_size,
                              hipStream_t stream) {
    const float* x     = (const float*)d_in[0];
    const int*   ei    = (const int*)d_in[1];
    const float* W1    = (const float*)d_in[2];
    const float* atts1 = (const float*)d_in[3];
    const float* attd1 = (const float*)d_in[4];
    const float* b1    = (const float*)d_in[5];
    const float* W2    = (const float*)d_in[6];
    const float* atts2 = (const float*)d_in[7];
    const float* attd2 = (const float*)d_in[8];
    const float* b2    = (const float*)d_in[9];
    const float* Wout  = (const float*)d_in[10];
    const float* bout  = (const float*)d_in[11];

    // Derive dims from in_sizes (reference: 50000,128,32,8,16,800000)
    const int HC1  = in_sizes[5];          // heads*Hd = 256
    const int Cd   = in_sizes[9];          // Hd = 32
    const int H1   = in_sizes[3] / Cd;     // 8
    const int Fin  = in_sizes[2] / HC1;    // 128
    const int N    = in_sizes[0] / Fin;    // 50000
    const int E    = in_sizes[1] / 2;      // 800000
    const int Fout = in_sizes[11];         // 16
    const int ET   = E + N;                // edges incl. self loops

    // Workspace layout (layer-2 buffers recycle layer-1 regions)
    char* ws = (char*)d_ws;
    size_t off = 0;
    auto alloc = [&](size_t bytes) {
        char* p = ws + off;
        off += (bytes + 255) & ~(size_t)255;
        return (float*)p;
    };
    float* Hraw1 = alloc((size_t)N * HC1 * 4);  // x@W1 ; reused for H2raw + Agg2
    float* Agg1  = alloc((size_t)N * HC1 * 4);  // layer1 aggregation -> h1act
    float* asrc1 = alloc((size_t)N * H1 * 4);
    float* adst1 = alloc((size_t)N * H1 * 4);
    float* nmax1 = alloc((size_t)N * H1 * 4);
    float* den1  = alloc((size_t)N * H1 * 4);
    // layer 2 reuses:
    float* H2raw = Hraw1;                       // [N, 32]
    float* Agg2  = Hraw1 + (size_t)N * Cd;      // [N, 32]
    float* asrc2 = asrc1;
    float* adst2 = adst1;
    float* nmax2 = nmax1;
    float* den2  = den1;

    const int TB = 256;

    // ---------------- Layer 1 ----------------
    {
        dim3 g(N / 16 + (N % 16 ? 1 : 0), (HC1 / 16 + 3) / 4);
        gemm_f32_wmma<false><<<g, 128, 0, stream>>>(x, W1, nullptr, Hraw1, N, HC1, Fin);
    }
    att_scores<<<(N * H1 + TB - 1) / TB, TB, 0, stream>>>(
        Hraw1, atts1, attd1, asrc1, adst1, N, H1, Cd);
    fill_f32<<<(N * H1 + TB - 1) / TB, TB, 0, stream>>>(nmax1, -INFINITY, N * H1);
    hipMemsetAsync(den1, 0, (size_t)N * H1 * 4, stream);
    hipMemsetAsync(Agg1, 0, (size_t)N * HC1 * 4, stream);
    edge_max<<<(ET + TB - 1) / TB, TB, 0, stream>>>(ei, E, N, H1, asrc1, adst1, nmax1);
    edge_expsum<<<(ET + TB - 1) / TB, TB, 0, stream>>>(ei, E, N, H1, asrc1, adst1, nmax1, den1);
    {
        int epb = TB / (H1 * Cd);  // = 1
        long long blocks = ((long long)ET + epb - 1) / epb;
        edge_aggregate<<<(unsigned)blocks, TB, 0, stream>>>(
            ei, E, N, H1, Cd, Hraw1, asrc1, adst1, nmax1, den1, Agg1);
    }
    bias_elu<<<((size_t)N * HC1 + TB - 1) / TB, TB, 0, stream>>>(Agg1, b1, N * HC1, HC1);

    // ---------------- Layer 2 (heads = 1) ----------------
    {
        dim3 g(N / 16 + (N % 16 ? 1 : 0), (Cd / 16 + 3) / 4);
        gemm_f32_wmma<false><<<g, 128, 0, stream>>>(Agg1, W2, nullptr, H2raw, N, Cd, HC1);
    }
    att_scores<<<(N + TB - 1) / TB, TB, 0, stream>>>(
        H2raw, atts2, attd2, asrc2, adst2, N, 1, Cd);
    fill_f32<<<(N + TB - 1) / TB, TB, 0, stream>>>(nmax2, -INFINITY, N);
    hipMemsetAsync(den2, 0, (size_t)N * 4, stream);
    hipMemsetAsync(Agg2, 0, (size_t)N * Cd * 4, stream);
    edge_max<<<(ET + TB - 1) / TB, TB, 0, stream>>>(ei, E, N, 1, asrc2, adst2, nmax2);
    edge_expsum<<<(ET + TB - 1) / TB, TB, 0, stream>>>(ei, E, N, 1, asrc2, adst2, nmax2, den2);
    {
        int epb = TB / Cd;  // = 8
        long long blocks = ((long long)ET + epb - 1) / epb;
        edge_aggregate<<<(unsigned)blocks, TB, 0, stream>>>(
            ei, E, N, 1, Cd, H2raw, asrc2, adst2, nmax2, den2, Agg2);
    }
    bias_elu<<<((size_t)N * Cd + TB - 1) / TB, TB, 0, stream>>>(Agg2, b2, N * Cd, Cd);

    // ---------------- Output projection ----------------
    {
        dim3 g(N / 16 + (N % 16 ? 1 : 0), (Fout / 16 + 3) / 4);
        gemm_f32_wmma<true><<<g, 128, 0, stream>>>(Agg2, Wout, bout, (float*)d_out,
                                                   N, Fout, Cd);
    }
}